// NeuralBestBuddies_16449724745315
// MI455X (gfx1250) — compile-verified
//
#include <hip/hip_runtime.h>
#include <hip/hip_bf16.h>
#include <stdint.h>

typedef __attribute__((ext_vector_type(2))) float v2f;
typedef __attribute__((ext_vector_type(8))) float v8f;
typedef unsigned int u32x4 __attribute__((ext_vector_type(4)));
typedef int i32x8 __attribute__((ext_vector_type(8)));
typedef int i32x4 __attribute__((ext_vector_type(4)));

#define C_CH   512
#define HDIM   64
#define WDIM   64
#define HW     4096
#define NIMG   2
#define GAMMA_ 0.05f

#define NCHUNK 144          // 9 offsets * (512/32) K-chunks

// ---- workspace layout (bytes, 256-aligned; Pn at +256 so TDM dj=-1 underrun stays in-bounds) ----
#define OFF_PN    ((size_t)256)
#define OFF_QN    ((size_t)16777472)
#define OFF_L2P   ((size_t)33554688)
#define OFF_L2Q   ((size_t)33587456)
#define OFF_PBEST ((size_t)33620224)
#define OFF_QBEST ((size_t)33685760)
#define OFF_MM    ((size_t)33751296)

#if defined(__gfx1250__) && __has_builtin(__builtin_amdgcn_tensor_load_to_lds)
#define USE_TDM 1
#else
#define USE_TDM 0
#endif

__device__ __forceinline__ unsigned orderF(float f) {
    unsigned u = __float_as_uint(f);
    return (u & 0x80000000u) ? ~u : (u | 0x80000000u);
}

// ---- reset packed-argmax buffers and norm min/max each call ----
__global__ void nbb_init_kernel(unsigned long long* __restrict__ Pbest,
                                unsigned long long* __restrict__ Qbest,
                                unsigned* __restrict__ mm) {
    int t = blockIdx.x * 256 + threadIdx.x;
    if (t < NIMG * HW)            Pbest[t] = 0ull;
    else if (t < 2 * NIMG * HW)   Qbest[t - NIMG * HW] = 0ull;
    if (t < 8) mm[t] = (t & 1) ? 0u : 0x7F800000u;   // even=min(+inf bits), odd=max(0)
}

// ---- per-pixel channel L2 norm, normalize, norm min/max (uint atomics on nonneg f32) ----
__global__ void nbb_norm_kernel(const float* __restrict__ Ps, const float* __restrict__ Qs,
                                float* __restrict__ Pn, float* __restrict__ Qn,
                                float* __restrict__ L2P, float* __restrict__ L2Q,
                                unsigned* __restrict__ mm) {
    int t = blockIdx.x * 256 + threadIdx.x;
    if (t >= NIMG * HW) return;
    int side = blockIdx.y;                 // 0 = P, 1 = Q
    int img = t / HW;
    const float* src = (side == 0 ? Ps : Qs) + (size_t)img * C_CH * HW + (t % HW);
    float* dst       = (side == 0 ? Pn : Qn) + (size_t)img * C_CH * HW + (t % HW);
    float sumsq = 0.0f;
    for (int c = 0; c < C_CH; ++c) {
        float v = src[(size_t)c * HW];
        sumsq += v * v;
    }
    float nrm = sqrtf(sumsq);
    (side == 0 ? L2P : L2Q)[t] = nrm;
    unsigned* slot = mm + 2 * (img * 2 + side);
    atomicMin(slot, __float_as_uint(nrm));
    atomicMax(slot + 1, __float_as_uint(nrm));
    float inv = 1.0f / nrm;
    for (int c = 0; c < C_CH; ++c)
        dst[(size_t)c * HW] = src[(size_t)c * HW] * inv;
}

#if USE_TDM
// Issue one 2D TDM tile load: rows x tileX f32 elements, row stride strideElems,
// LDS padding: pad_amount=4 dwords after every 2^(padCode+1) dwords (ISA D# groups 0/1).
__device__ __forceinline__ void tdm_load_2d(unsigned ldsByteAddr, const float* gaddr,
                                            unsigned tileX, unsigned rows,
                                            unsigned strideElems, unsigned padCode) {
    unsigned long long ga = (unsigned long long)(uintptr_t)gaddr;
    u32x4 g0;
    g0[0] = 1u;                                            // count=1, user mode
    g0[1] = ldsByteAddr;                                   // lds_addr
    g0[2] = (unsigned)(ga & 0xFFFFFFFFull);                // global_addr[31:0]
    g0[3] = (unsigned)((ga >> 32) & 0x1FFFFFFull) | (2u << 30);  // addr[56:32] | type=2
    const unsigned dim0 = 0x40000000u;                     // huge tensor dims: no HW OOB masking
    const unsigned dim1 = rows;
    i32x8 g1;
    g1[0] = (int)((2u << 16) | (1u << 20) | (padCode << 22) | (3u << 25)); // 4B elems, pad_en, +4dw
    g1[1] = (int)((dim0 & 0xFFFFu) << 16);
    g1[2] = (int)(((dim0 >> 16) & 0xFFFFu) | ((dim1 & 0xFFFFu) << 16));
    g1[3] = (int)(((dim1 >> 16) & 0xFFFFu) | (tileX << 16));               // tile_dim0
    g1[4] = (int)(rows & 0xFFFFu);                                         // tile_dim1 (dim2=0)
    g1[5] = (int)strideElems;                                              // dim0_stride lo
    g1[6] = 0;
    g1[7] = 0;
    i32x4 z4 = {0, 0, 0, 0};
    i32x8 z8 = {0, 0, 0, 0, 0, 0, 0, 0};
    // clang-23 / amdgpu-toolchain form: 6 args (g0, g1, g2, g3, extra group, cpol)
    __builtin_amdgcn_tensor_load_to_lds(g0, g1, z4, z4, z8, 0);
}
#endif

// ---- fused cost-volume GEMM (K = 9*512 via shifted/masked tiles) + tile argmax ----
// 64(p) x 128(q) Cvol tile per workgroup; 8 waves in 2(M)x4(N) grid, 4 accumulators/wave.
// TDM path: double-buffered LDS tiles filled by tensor_load_to_lds, zero-fix for padding edges,
// chunk t+1 DMA overlaps chunk t WMMAs; TENSORcnt + barriers for sync.
__launch_bounds__(256)
__global__ void nbb_cost_argmax_kernel(const float* __restrict__ Pn, const float* __restrict__ Qn,
                                       unsigned long long* __restrict__ Pbest,
                                       unsigned long long* __restrict__ Qbest) {
    // pool: A0[32*68] A1[32*68] B0[32*132] B1[32*132]; C tile (64*129) aliases the front.
    __shared__ float pool[12800];
#define A_OFF(b) ((b) ? 2176u : 0u)
#define B_OFF(b) (4352u + ((b) ? 4224u : 0u))

    const int img = blockIdx.z;
    const int ip  = blockIdx.y;            // p-tile = image row ip
    const int iq  = blockIdx.x;            // q-tile = image rows 2iq, 2iq+1

    const float* Pni = Pn + (size_t)img * C_CH * HW;
    const float* Qni = Qn + (size_t)img * C_CH * HW;

    const int tid   = threadIdx.x;
    const int lane  = tid & 31;            // wave32
    const int wave  = tid >> 5;
    const int laneM = lane & 15;
    const int half  = lane >> 4;
    const int m0    = (wave & 1) * 32;
    const int n0    = (wave >> 1) * 32;

    v8f acc00 = {0.f,0.f,0.f,0.f,0.f,0.f,0.f,0.f};
    v8f acc01 = acc00, acc10 = acc00, acc11 = acc00;

#if USE_TDM
    const unsigned ldsBase = (unsigned)(uintptr_t)(void*)pool;  // low 32 bits = LDS byte address
    auto issue_chunk = [&](int t, int b) {
        const int d  = t >> 4;
        const int kc = (t & 15) * 32;
        const int di = d / 3 - 1;
        const int dj = d % 3 - 1;
        int prow = ip + di;
        if (prow < 0) prow = 0; else if (prow > 63) prow = 63;   // clamp; invalid tile zeroed later
        tdm_load_2d(ldsBase + 4u * A_OFF(b), Pni + (size_t)kc * HW + prow * WDIM + dj,
                    64u, 32u, (unsigned)HW, 5u);                 // pad 4dw per 64dw -> stride 68
        const int qb = (iq * 2 + di) * WDIM + dj;                // contiguous 128-wide row
        tdm_load_2d(ldsBase + 4u * B_OFF(b), Qni + (size_t)kc * HW + qb,
                    128u, 32u, (unsigned)HW, 6u);                // pad 4dw per 128dw -> stride 132
    };

    if (wave == 0) issue_chunk(0, 0);
    for (int t = 0; t < NCHUNK; ++t) {
        const int b = t & 1;
        if (wave == 0) __builtin_amdgcn_s_wait_tensorcnt((short)0);
        __syncthreads();                                   // tiles t ready; buffer b^1 free
        if (wave == 0 && t + 1 < NCHUNK) issue_chunk(t + 1, b ^ 1);

        // zero-fix the DMA'd tiles (padding semantics of the reference)
        float* A = pool + A_OFF(b);
        float* B = pool + B_OFF(b);
        const int d  = t >> 4;
        const int di = d / 3 - 1;
        const int dj = d % 3 - 1;
        const int prow = ip + di;
        if ((unsigned)prow >= (unsigned)HDIM) {
            for (int e = tid; e < 2048; e += 256) A[(e >> 6) * 68 + (e & 63)] = 0.0f;
        } else if (dj != 0 && tid < 32) {
            A[tid * 68 + (dj < 0 ? 0 : 63)] = 0.0f;
        }
        const int qr0 = iq * 2 + di;
        if (qr0 < 0)
            for (int e = tid; e < 2048; e += 256) B[(e >> 6) * 132 + (e & 63)] = 0.0f;
        if (qr0 + 1 > 63)
            for (int e = tid; e < 2048; e += 256) B[(e >> 6) * 132 + 64 + (e & 63)] = 0.0f;
        if (dj != 0 && tid < 32) {
            const int c = (dj < 0 ? 0 : 63);
            B[tid * 132 + c] = 0.0f;
            B[tid * 132 + 64 + c] = 0.0f;
        }
        __syncthreads();
#else
    float rA[8], rB[16];
    auto load_chunk = [&](int t) {
        const int d  = t >> 4;
        const int kc = (t & 15) * 32;
        const int di = d / 3 - 1;
        const int dj = d % 3 - 1;
        const int prow = ip + di;
        const bool prowOK = (unsigned)prow < (unsigned)HDIM;
#pragma unroll
        for (int s = 0; s < 8; ++s) {
            const int e = tid + s * 256, k = e >> 6, m = e & 63, jj = m + dj;
            float v = 0.0f;
            if (prowOK && (unsigned)jj < (unsigned)WDIM)
                v = Pni[(size_t)(kc + k) * HW + prow * WDIM + jj];
            rA[s] = v;
        }
#pragma unroll
        for (int s = 0; s < 16; ++s) {
            const int e = tid + s * 256, k = e >> 7, n = e & 127;
            const int qr = iq * 2 + (n >> 6) + di, jj = (n & 63) + dj;
            float v = 0.0f;
            if ((unsigned)qr < (unsigned)HDIM && (unsigned)jj < (unsigned)WDIM)
                v = Qni[(size_t)(kc + k) * HW + qr * WDIM + jj];
            rB[s] = v;
        }
    };
    load_chunk(0);
    for (int t = 0; t < NCHUNK; ++t) {
        const int b = 0;
        float* A = pool + A_OFF(0);
        float* B = pool + B_OFF(0);
        __syncthreads();
#pragma unroll
        for (int s = 0; s < 8; ++s) {
            const int e = tid + s * 256;
            A[(e >> 6) * 68 + (e & 63)] = rA[s];
        }
#pragma unroll
        for (int s = 0; s < 16; ++s) {
            const int e = tid + s * 256;
            B[(e >> 7) * 132 + (e & 127)] = rB[s];
        }
        __syncthreads();
        if (t + 1 < NCHUNK) load_chunk(t + 1);
#endif
        {
            float* Ab = pool + A_OFF(b);
            float* Bb = pool + B_OFF(b);
            // 8 K-steps x 4 WMMAs (V_WMMA_F32_16X16X4_F32)
#pragma unroll
            for (int kk = 0; kk < 32; kk += 4) {
                const int kb = kk + 2 * half;      // lane half selects K pair (ISA A layout)
                v2f a0, a1, b0, b1;
                a0.x = Ab[kb * 68 + m0 + laneM];
                a0.y = Ab[(kb + 1) * 68 + m0 + laneM];
                a1.x = Ab[kb * 68 + m0 + 16 + laneM];
                a1.y = Ab[(kb + 1) * 68 + m0 + 16 + laneM];
                b0.x = Bb[kb * 132 + n0 + laneM];
                b0.y = Bb[(kb + 1) * 132 + n0 + laneM];
                b1.x = Bb[kb * 132 + n0 + 16 + laneM];
                b1.y = Bb[(kb + 1) * 132 + n0 + 16 + laneM];
                acc00 = __builtin_amdgcn_wmma_f32_16x16x4_f32(false, a0, false, b0,
                                                              (short)0, acc00, false, false);
                acc01 = __builtin_amdgcn_wmma_f32_16x16x4_f32(false, a0, false, b1,
                                                              (short)0, acc01, false, false);
                acc10 = __builtin_amdgcn_wmma_f32_16x16x4_f32(false, a1, false, b0,
                                                              (short)0, acc10, false, false);
                acc11 = __builtin_amdgcn_wmma_f32_16x16x4_f32(false, a1, false, b1,
                                                              (short)0, acc11, false, false);
            }
        }
    }

    __syncthreads();
    // C tile aliases the pool (K-loop done): spill accumulators, stride 129 (odd -> conflict-free)
    float* Cs = pool;
#pragma unroll
    for (int r = 0; r < 8; ++r) {
        const int r0 = m0 + r + 8 * half;
        const int r1 = r0 + 16;
        Cs[r0 * 129 + n0 + laneM]      = acc00[r];
        Cs[r0 * 129 + n0 + 16 + laneM] = acc01[r];
        Cs[r1 * 129 + n0 + laneM]      = acc10[r];
        Cs[r1 * 129 + n0 + 16 + laneM] = acc11[r];
    }
    __syncthreads();

    if (tid < 64) {                         // per-p argmax over this q-tile (128 cols)
        const int p = tid;
        float best = Cs[p * 129];
        int bq = 0;
        for (int q = 1; q < 128; ++q) {
            float v = Cs[p * 129 + q];
            if (v > best) { best = v; bq = q; }
        }
        unsigned long long key = ((unsigned long long)orderF(best) << 32)
                               | (unsigned long long)(0xFFFFFFFFu - (unsigned)(iq * 128 + bq));
        atomicMax(&Pbest[img * HW + ip * 64 + p], key);
    } else if (tid < 192) {                 // per-q argmax over this p-tile (64 rows)
        const int q = tid - 64;
        float best = Cs[q];
        int bp = 0;
        for (int p = 1; p < 64; ++p) {
            float v = Cs[p * 129 + q];
            if (v > best) { best = v; bp = p; }
        }
        unsigned long long key = ((unsigned long long)orderF(best) << 32)
                               | (unsigned long long)(0xFFFFFFFFu - (unsigned)(ip * 64 + bp));
        atomicMax(&Qbest[img * HW + iq * 128 + q], key);
    }
}

// ---- decode packed argmax, compute best-buddy mask, write outputs as float ----
__global__ void nbb_finalize_kernel(const unsigned long long* __restrict__ Pbest,
                                    const unsigned long long* __restrict__ Qbest,
                                    const float* __restrict__ L2P, const float* __restrict__ L2Q,
                                    const unsigned* __restrict__ mm,
                                    float* __restrict__ out) {
    int t = blockIdx.x * 256 + threadIdx.x;
    if (t >= NIMG * HW) return;
    const int img = t / HW;
    const int p = t % HW;
    const unsigned pq = 0xFFFFFFFFu - (unsigned)(Pbest[t] & 0xFFFFFFFFull);  // P_nearest[p]
    const unsigned qp = 0xFFFFFFFFu - (unsigned)(Qbest[t] & 0xFFFFFFFFull);  // Q_nearest[q]
    out[NIMG * HW + t]     = (float)pq;
    out[2 * NIMG * HW + t] = (float)qp;
    const unsigned qn_of_pq = 0xFFFFFFFFu - (unsigned)(Qbest[img * HW + pq] & 0xFFFFFFFFull);
    const float pmin = __uint_as_float(mm[2 * (img * 2 + 0) + 0]);
    const float pmax = __uint_as_float(mm[2 * (img * 2 + 0) + 1]);
    const float qmin = __uint_as_float(mm[2 * (img * 2 + 1) + 0]);
    const float qmax = __uint_as_float(mm[2 * (img * 2 + 1) + 1]);
    const float pN = (L2P[t] - pmin) / (pmax - pmin);
    const float qN = (L2Q[img * HW + pq] - qmin) / (qmax - qmin);
    const bool bb = (qn_of_pq == (unsigned)p) && (pN > GAMMA_) && (qN > GAMMA_);
    out[t] = bb ? 1.0f : 0.0f;
}

extern "C" void kernel_launch(void* const* d_in, const int* in_sizes, int n_in,
                              void* d_out, int out_size, void* d_ws, size_t ws_size,
                              hipStream_t stream) {
    const float* Ps = (const float*)d_in[0];
    const float* Qs = (const float*)d_in[1];
    // d_in[2] = neigh_rad (always 1; the 3x3 window is compiled in, matching NEIGH_RAD_STATIC)

    char* ws = (char*)d_ws;
    float* Pn  = (float*)(ws + OFF_PN);
    float* Qn  = (float*)(ws + OFF_QN);
    float* L2P = (float*)(ws + OFF_L2P);
    float* L2Q = (float*)(ws + OFF_L2Q);
    unsigned long long* Pbest = (unsigned long long*)(ws + OFF_PBEST);
    unsigned long long* Qbest = (unsigned long long*)(ws + OFF_QBEST);
    unsigned* mm = (unsigned*)(ws + OFF_MM);
    float* out = (float*)d_out;

    nbb_init_kernel<<<(2 * NIMG * HW + 255) / 256, 256, 0, stream>>>(Pbest, Qbest, mm);
    nbb_norm_kernel<<<dim3((NIMG * HW + 255) / 256, 2), 256, 0, stream>>>(
        Ps, Qs, Pn, Qn, L2P, L2Q, mm);
    nbb_cost_argmax_kernel<<<dim3(HDIM / 2, HDIM, NIMG), 256, 0, stream>>>(Pn, Qn, Pbest, Qbest);
    nbb_finalize_kernel<<<(NIMG * HW + 255) / 256, 256, 0, stream>>>(
        Pbest, Qbest, L2P, L2Q, mm, out);
}